// HNNLayer_7662221656116
// MI455X (gfx1250) — compile-verified
//
#include <hip/hip_runtime.h>

// ---------------------------------------------------------------------------
// HNN layer for MI455X (gfx1250, wave32).
// Dense GEMMs -> V_WMMA_F32_16X16X4_F32, 64x128 block tile, 4 accumulators per
// wave so each B fragment (W tile, cache-resident) feeds 4 WMMAs.
// Segment-means / SpMM -> wave-per-row kernels: 32 lanes x float4 = 128 cols,
// coalesced 512B row reads + global_atomic_add_f32 scatter.
// ---------------------------------------------------------------------------

typedef __attribute__((ext_vector_type(2))) float v2f;
typedef __attribute__((ext_vector_type(8))) float v8f;

#define DD 128
#define SX_PITCH 132   // pitch mod 64 banks = 4 -> ds_load_b64 hits all 64 banks
#define ROWS_PER_BLK 64

// Y[r][c] = act( sum_k X[r][k] * W[c*ldw + koff + k] ),  X:[R,128], output [R,128]
__launch_bounds__(256)
__global__ void gemm_xwT_wmma(const float* __restrict__ X, int R,
                              const float* __restrict__ W, int ldw, int koff,
                              float* __restrict__ Y, int do_relu)
{
    __shared__ float sX[ROWS_PER_BLK * SX_PITCH];

    const int t = threadIdx.x;
    const int rowBase = blockIdx.x * ROWS_PER_BLK;

    // Stage the 64x128 X strip (coalesced float4 global loads, 8 per thread).
    for (int i = t; i < ROWS_PER_BLK * 32; i += 256) {
        const int m = i >> 5;            // row in strip
        const int k = (i & 31) << 2;     // column (float4 granularity)
        const int r = rowBase + m;
        float4 x = (r < R) ? *(const float4*)(X + (size_t)r * DD + k)
                           : make_float4(0.f, 0.f, 0.f, 0.f);
        *(float4*)(sX + m * SX_PITCH + k) = x;
    }
    __syncthreads();

    const int wave  = t >> 5;            // 0..7 -> 16-column tile
    const int lane  = t & 31;
    const int cTile = wave * 16;
    const int mr    = lane & 15;         // A-row / B-col index for this lane
    const int khalf = (lane >> 4) * 2;   // lanes 16..31 carry K = k0+2, k0+3

    const float* wcol = W + (size_t)(cTile + mr) * ldw + koff;  // cache-resident

    v8f acc[4] = {{}, {}, {}, {}};

    #pragma unroll
    for (int k0 = 0; k0 < DD; k0 += 4) {
        v2f b;                            // B fragment: loaded once, used 4x
        b.x = wcol[k0 + khalf];
        b.y = wcol[k0 + khalf + 1];
        #pragma unroll
        for (int rt = 0; rt < 4; ++rt) {
            const float* xrow = sX + (mr + 16 * rt) * SX_PITCH;
            v2f a;
            a.x = xrow[k0 + khalf];
            a.y = xrow[k0 + khalf + 1];
            acc[rt] = __builtin_amdgcn_wmma_f32_16x16x4_f32(
                /*neg_a=*/false, a, /*neg_b=*/false, b,
                /*c_mod=*/(short)0, acc[rt], /*reuse_a=*/false, /*reuse_b=*/false);
        }
    }

    // C/D layout: VGPR i holds row (i + 8*(lane>=16)), col = cTile + (lane&15).
    const int col  = cTile + (lane & 15);
    const int rOff = (lane >> 4) * 8;
    #pragma unroll
    for (int rt = 0; rt < 4; ++rt) {
        #pragma unroll
        for (int i = 0; i < 8; i++) {
            const int r = rowBase + rt * 16 + rOff + i;
            if (r < R) {
                float v = acc[rt][i];
                if (do_relu) v = fmaxf(v, 0.0f);
                Y[(size_t)r * DD + col] = v;
            }
        }
    }
}

// ---------------------------------------------------------------------------

__global__ void fill_kernel(float* __restrict__ p, size_t n)
{
    size_t i = (size_t)blockIdx.x * blockDim.x + threadIdx.x;
    const size_t stride = (size_t)gridDim.x * blockDim.x;
    for (; i < n; i += stride) p[i] = 0.0f;
}

__global__ void count_kernel(const int* __restrict__ iv, const int* __restrict__ ie,
                             float* __restrict__ cv, float* __restrict__ ce, int E)
{
    const int j = blockIdx.x * blockDim.x + threadIdx.x;
    if (j < E) {
        atomicAdd(cv + iv[j], 1.0f);
        atomicAdd(ce + ie[j], 1.0f);
    }
}

// sum[ie[j]] += (pv[iv[j]] + pe[ie[j]] + bias) * (scale ? scale[iv[j]] : 1)
__global__ void scatter_psi_kernel(const float* __restrict__ pv, const float* __restrict__ pe,
                                   const float* __restrict__ bias, const float* __restrict__ scale,
                                   const int* __restrict__ iv, const int* __restrict__ ie,
                                   float* __restrict__ sum, int E)
{
    const long long tid = (long long)blockIdx.x * blockDim.x + threadIdx.x;
    const int j = (int)(tid >> 5);
    const int q = (int)(tid & 31);
    if (j >= E) return;
    const int v = iv[j], e = ie[j];
    const float s = scale ? scale[v] : 1.0f;
    const int c = q * 4;
    const float4 a  = *(const float4*)(pv + (size_t)v * DD + c);
    const float4 b  = *(const float4*)(pe + (size_t)e * DD + c);
    const float4 bb = *(const float4*)(bias + c);
    float* dst = sum + (size_t)e * DD + c;
    atomicAdd(dst + 0, (a.x + b.x + bb.x) * s);
    atomicAdd(dst + 1, (a.y + b.y + bb.y) * s);
    atomicAdd(dst + 2, (a.z + b.z + bb.z) * s);
    atomicAdd(dst + 3, (a.w + b.w + bb.w) * s);
}

// sum[sidx[j]] += src[gidx[j]]
__global__ void gather_scatter_kernel(const float* __restrict__ src,
                                      const int* __restrict__ gidx, const int* __restrict__ sidx,
                                      float* __restrict__ sum, int E)
{
    const long long tid = (long long)blockIdx.x * blockDim.x + threadIdx.x;
    const int j = (int)(tid >> 5);
    const int q = (int)(tid & 31);
    if (j >= E) return;
    const int g = gidx[j], s = sidx[j];
    const int c = q * 4;
    const float4 x = *(const float4*)(src + (size_t)g * DD + c);
    float* dst = sum + (size_t)s * DD + c;
    atomicAdd(dst + 0, x.x);
    atomicAdd(dst + 1, x.y);
    atomicAdd(dst + 2, x.z);
    atomicAdd(dst + 3, x.w);
}

// out[row[j]] += val[j] * x[col[j]]
__global__ void spmm_kernel(const int* __restrict__ row, const int* __restrict__ col,
                            const float* __restrict__ val, const float* __restrict__ x,
                            float* __restrict__ out, int nnz)
{
    const long long tid = (long long)blockIdx.x * blockDim.x + threadIdx.x;
    const int j = (int)(tid >> 5);
    const int q = (int)(tid & 31);
    if (j >= nnz) return;
    const int r = row[j], cl = col[j];
    const float v = val[j];
    const int c = q * 4;
    const float4 xv = *(const float4*)(x + (size_t)cl * DD + c);
    float* dst = out + (size_t)r * DD + c;
    atomicAdd(dst + 0, v * xv.x);
    atomicAdd(dst + 1, v * xv.y);
    atomicAdd(dst + 2, v * xv.z);
    atomicAdd(dst + 3, v * xv.w);
}

// out[r] = sum[r] / max(cnt[r],1) (+ addend[r])
__global__ void seg_div_kernel(const float* __restrict__ sum, const float* __restrict__ cnt,
                               const float* __restrict__ addend, float* __restrict__ out, int rows)
{
    const long long tid = (long long)blockIdx.x * blockDim.x + threadIdx.x;
    const int r = (int)(tid >> 5);
    const int q = (int)(tid & 31);
    if (r >= rows) return;
    const int c = q * 4;
    const float inv = 1.0f / fmaxf(cnt[r], 1.0f);
    float4 s = *(const float4*)(sum + (size_t)r * DD + c);
    s.x *= inv; s.y *= inv; s.z *= inv; s.w *= inv;
    if (addend) {
        const float4 a = *(const float4*)(addend + (size_t)r * DD + c);
        s.x += a.x; s.y += a.y; s.z += a.z; s.w += a.w;
    }
    *(float4*)(out + (size_t)r * DD + c) = s;
}

// out = a + b (elementwise, float4)
__global__ void add_kernel(const float* __restrict__ a, const float* __restrict__ b,
                           float* __restrict__ out, size_t n4)
{
    const size_t i = (size_t)blockIdx.x * blockDim.x + threadIdx.x;
    if (i >= n4) return;
    const float4 av = ((const float4*)a)[i];
    const float4 bv = ((const float4*)b)[i];
    ((float4*)out)[i] = make_float4(av.x + bv.x, av.y + bv.y, av.z + bv.z, av.w + bv.w);
}

// out[r] = x[r] * s[r]
__global__ void scale_rows_kernel(const float* __restrict__ x, const float* __restrict__ s,
                                  float* __restrict__ out, int rows)
{
    const long long tid = (long long)blockIdx.x * blockDim.x + threadIdx.x;
    const int r = (int)(tid >> 5);
    const int q = (int)(tid & 31);
    if (r >= rows) return;
    const int c = q * 4;
    const float sv = s[r];
    float4 xv = *(const float4*)(x + (size_t)r * DD + c);
    xv.x *= sv; xv.y *= sv; xv.z *= sv; xv.w *= sv;
    *(float4*)(out + (size_t)r * DD + c) = xv;
}

// ---------------------------------------------------------------------------

static inline int cdiv_ll(long long a, long long b) { return (int)((a + b - 1) / b); }

extern "C" void kernel_launch(void* const* d_in, const int* in_sizes, int n_in,
                              void* d_out, int out_size, void* d_ws, size_t ws_size,
                              hipStream_t stream)
{
    (void)n_in; (void)out_size; (void)ws_size;
    const int D = DD;
    const float* vfeat  = (const float*)d_in[0];
    const float* efeat  = (const float*)d_in[1];
    const float* invDV  = (const float*)d_in[2];
    const float* invDE  = (const float*)d_in[3];
    const int*   inc_v  = (const int*)d_in[4];
    const int*   inc_e  = (const int*)d_in[5];
    const int*   eRow   = (const int*)d_in[6];
    const int*   eCol   = (const int*)d_in[7];
    const float* eVal   = (const float*)d_in[8];
    const int*   vRow   = (const int*)d_in[9];
    const int*   vCol   = (const int*)d_in[10];
    const float* vVal   = (const float*)d_in[11];
    const float* W_psi1 = (const float*)d_in[12];
    const float* b_psi1 = (const float*)d_in[13];
    const float* W_psi2 = (const float*)d_in[14];
    const float* b_psi2 = (const float*)d_in[15];
    const float* W_v    = (const float*)d_in[16];
    const float* W_e    = (const float*)d_in[17];

    const int N    = in_sizes[0] / D;
    const int M    = in_sizes[1] / D;
    const int E    = in_sizes[4];
    const int nnzE = in_sizes[6];
    const int nnzV = in_sizes[9];

    // Workspace layout (floats): 2*N*D + 3*M*D + N + M  (~134 MB)
    float* ws = (float*)d_ws;
    size_t o = 0;
    float* bufA_N = ws + o; o += (size_t)N * D;
    float* bufB_N = ws + o; o += (size_t)N * D;
    float* bufA_M = ws + o; o += (size_t)M * D;
    float* bufB_M = ws + o; o += (size_t)M * D;
    float* bufC_M = ws + o; o += (size_t)M * D;
    float* cnt_v  = ws + o; o += (size_t)N;
    float* cnt_e  = ws + o; o += (size_t)M;

    float* vout = (float*)d_out;                       // vfeat_new [N,D]
    float* eout = (float*)d_out + (size_t)N * D;       // efeat_new [M,D]

    const int TB = 256;
    const int gN  = cdiv_ll((long long)N * 32, TB);    // wave-per-row grids
    const int gM  = cdiv_ll((long long)M * 32, TB);
    const int gE  = cdiv_ll((long long)E * 32, TB);
    const int gZE = cdiv_ll((long long)nnzE * 32, TB);
    const int gZV = cdiv_ll((long long)nnzV * 32, TB);
    const int gGN = (N + ROWS_PER_BLK - 1) / ROWS_PER_BLK;   // GEMM grids
    const int gGM = (M + ROWS_PER_BLK - 1) / ROWS_PER_BLK;

    // --- counts ---------------------------------------------------------
    fill_kernel<<<1024, TB, 0, stream>>>(cnt_v, (size_t)N);
    fill_kernel<<<1024, TB, 0, stream>>>(cnt_e, (size_t)M);
    count_kernel<<<cdiv_ll(E, TB), TB, 0, stream>>>(inc_v, inc_e, cnt_v, cnt_e, E);

    // --- pv1 = vfeat @ W1v^T, pe1 = efeat @ W1e^T -----------------------
    gemm_xwT_wmma<<<gGN, TB, 0, stream>>>(vfeat, N, W_psi1, 2 * D, 0, bufA_N, 0);
    gemm_xwT_wmma<<<gGM, TB, 0, stream>>>(efeat, M, W_psi1, 2 * D, D, bufA_M, 0);

    // --- A = seg_mean(psi1, inc_e) --------------------------------------
    fill_kernel<<<1024, TB, 0, stream>>>(bufB_M, (size_t)M * D);
    scatter_psi_kernel<<<gE, TB, 0, stream>>>(bufA_N, bufA_M, b_psi1, invDV, inc_v, inc_e, bufB_M, E);
    seg_div_kernel<<<gM, TB, 0, stream>>>(bufB_M, cnt_e, nullptr, bufB_M, M);

    // --- A = spmm(eMat, A); _efeat = A + efeat --------------------------
    fill_kernel<<<1024, TB, 0, stream>>>(bufC_M, (size_t)M * D);
    spmm_kernel<<<gZE, TB, 0, stream>>>(eRow, eCol, eVal, bufB_M, bufC_M, nnzE);
    add_kernel<<<cdiv_ll((long long)M * D / 4, TB), TB, 0, stream>>>(bufC_M, efeat, bufC_M, (size_t)M * D / 4);

    // --- _vfeat = seg_mean(_efeat[inc_e], inc_v) ------------------------
    fill_kernel<<<1024, TB, 0, stream>>>(bufB_N, (size_t)N * D);
    gather_scatter_kernel<<<gE, TB, 0, stream>>>(bufC_M, inc_e, inc_v, bufB_N, E);
    seg_div_kernel<<<gN, TB, 0, stream>>>(bufB_N, cnt_v, nullptr, bufB_N, N);

    // --- vfeat_new = relu(_vfeat @ W_v^T) -------------------------------
    gemm_xwT_wmma<<<gGN, TB, 0, stream>>>(bufB_N, N, W_v, D, 0, vout, 1);

    // --- psi2 = (vfeat_new @ W2v^T)[iv] + (efeat @ W2e^T)[ie] + b2 ------
    gemm_xwT_wmma<<<gGN, TB, 0, stream>>>(vout,  N, W_psi2, 2 * D, 0, bufA_N, 0);
    gemm_xwT_wmma<<<gGM, TB, 0, stream>>>(efeat, M, W_psi2, 2 * D, D, bufA_M, 0);
    fill_kernel<<<1024, TB, 0, stream>>>(bufB_M, (size_t)M * D);
    scatter_psi_kernel<<<gE, TB, 0, stream>>>(bufA_N, bufA_M, b_psi2, nullptr, inc_v, inc_e, bufB_M, E);
    seg_div_kernel<<<gM, TB, 0, stream>>>(bufB_M, cnt_e, nullptr, bufB_M, M);   // B

    // --- _ef2 = efeat * invDE; _vf2 = seg_mean(_ef2[inc_e], inc_v) ------
    scale_rows_kernel<<<gM, TB, 0, stream>>>(efeat, invDE, bufC_M, M);
    fill_kernel<<<1024, TB, 0, stream>>>(bufB_N, (size_t)N * D);
    gather_scatter_kernel<<<gE, TB, 0, stream>>>(bufC_M, inc_e, inc_v, bufB_N, E);
    seg_div_kernel<<<gN, TB, 0, stream>>>(bufB_N, cnt_v, nullptr, bufB_N, N);

    // --- _vf2 = spmm(vMat, _vf2) ----------------------------------------
    fill_kernel<<<1024, TB, 0, stream>>>(bufA_N, (size_t)N * D);
    spmm_kernel<<<gZV, TB, 0, stream>>>(vRow, vCol, vVal, bufB_N, bufA_N, nnzV);

    // --- _ef3 = seg_mean(_vf2[inc_v], inc_e) + B ------------------------
    fill_kernel<<<1024, TB, 0, stream>>>(bufA_M, (size_t)M * D);
    gather_scatter_kernel<<<gE, TB, 0, stream>>>(bufA_N, inc_v, inc_e, bufA_M, E);
    seg_div_kernel<<<gM, TB, 0, stream>>>(bufA_M, cnt_e, bufB_M, bufA_M, M);

    // --- efeat_new = relu(_ef3 @ W_e^T) ---------------------------------
    gemm_xwT_wmma<<<gGM, TB, 0, stream>>>(bufA_M, M, W_e, D, 0, eout, 1);
}